// KS_8134668058856
// MI455X (gfx1250) — compile-verified
//
#include <hip/hip_runtime.h>

// KS statistic: histogram split by label -> cumsum -> max |CDF_pos - CDF_neg|
// Phase 1: LDS-privatized packed histogram (bandwidth bound, ~11us floor @ 23.3TB/s)
// Phase 2: single-wave WMMA triangular-matrix scan (v_wmma_f32_16x16x4_f32)

#define NBINS   10001
#define PADBINS 10240          // 40 tiles * 256 bins, zero-padded
#define NTILES  (PADBINS / 256)

typedef __attribute__((ext_vector_type(2))) float v2f;
typedef __attribute__((ext_vector_type(8))) float v8f;

__global__ void ks_zero(unsigned* __restrict__ ws) {
    int i = blockIdx.x * blockDim.x + threadIdx.x;
    if (i < 2 * PADBINS) ws[i] = 0u;
}

__global__ __launch_bounds__(256) void ks_hist(const float* __restrict__ preds,
                                               const float* __restrict__ targets,
                                               unsigned* __restrict__ g_tp,
                                               unsigned* __restrict__ g_fp,
                                               int n) {
    // packed per-block histogram: pos count in [31:16], neg count in [15:0]
    __shared__ unsigned lhist[NBINS];
    const int tid = threadIdx.x;
    for (int b = tid; b < NBINS; b += blockDim.x) lhist[b] = 0u;
    __syncthreads();

    const int n4 = n >> 2;
    const long long T = (long long)gridDim.x * blockDim.x;
    const float4* p4 = (const float4*)preds;
    const float4* t4 = (const float4*)targets;
    for (long long i = (long long)blockIdx.x * blockDim.x + tid; i < n4; i += T) {
        float4 x = p4[i];
        float4 y = t4[i];
        float xs[4] = {x.x, x.y, x.z, x.w};
        float ys[4] = {y.x, y.y, y.z, y.w};
#pragma unroll
        for (int k = 0; k < 4; ++k) {
            float s = 1.0f / (1.0f + __expf(-xs[k]));   // sigmoid via v_exp_f32
            int bin = (int)(10000.0f * s);
            bin = bin < 0 ? 0 : (bin > 10000 ? 10000 : bin);
            unsigned val = (ys[k] >= 0.5f) ? 0x10000u : 1u;
            atomicAdd(&lhist[bin], val);                 // ds_add_u32
        }
    }
    // scalar tail (n not multiple of 4)
    if (blockIdx.x == 0 && tid == 0) {
        for (int i = n4 << 2; i < n; ++i) {
            float s = 1.0f / (1.0f + __expf(-preds[i]));
            int bin = (int)(10000.0f * s);
            bin = bin < 0 ? 0 : (bin > 10000 ? 10000 : bin);
            atomicAdd(&lhist[bin], (targets[i] >= 0.5f) ? 0x10000u : 1u);
        }
    }
    __syncthreads();
    // flush to global (unpacked)
    for (int b = tid; b < NBINS; b += blockDim.x) {
        unsigned v = lhist[b];
        if (v) {
            unsigned hi = v >> 16, lo = v & 0xFFFFu;
            if (hi) atomicAdd(&g_tp[b], hi);
            if (lo) atomicAdd(&g_fp[b], lo);
        }
    }
}

// Single wave32. Inclusive scan of diff[] via D = L*M with L = lower-triangular ones,
// chained as four V_WMMA_F32_16X16X4_F32 K-slices per 16x16 tile (256 bins/tile).
__global__ __launch_bounds__(32) void ks_scan(const unsigned* __restrict__ g_tp,
                                              const unsigned* __restrict__ g_fp,
                                              float* __restrict__ out) {
    const int lane = threadIdx.x;          // wave32
    // class totals
    unsigned sp = 0u, sn = 0u;
    for (int b = lane; b < NBINS; b += 32) { sp += g_tp[b]; sn += g_fp[b]; }
    for (int off = 16; off; off >>= 1) {
        sp += __shfl_xor(sp, off);
        sn += __shfl_xor(sn, off);
    }
    const float invP = 1.0f / (float)sp;
    const float invN = 1.0f / (float)sn;

    const int c    = lane & 15;   // column (B/C/D lane striping)
    const int half = lane >> 4;   // 0: K/M rows {0,1}/{0..7}; 1: {2,3}/{8..15}
    const int m    = lane & 15;   // A-matrix row for this lane

    float carry = 0.0f;           // cumsum carry across tiles
    float mx    = 0.0f;

    for (int tile = 0; tile < NTILES; ++tile) {
        const int base = tile * 256;
        v8f acc = {0.f, 0.f, 0.f, 0.f, 0.f, 0.f, 0.f, 0.f};
#pragma unroll
        for (int kblk = 0; kblk < 4; ++kblk) {
            const int k0 = 4 * kblk + 2 * half;   // global K of reg0 (reg1 = k0+1)
            v2f a, b;
            // L[m][k] = (m >= k)
            a.x = (m >= k0)     ? 1.0f : 0.0f;
            a.y = (m >= k0 + 1) ? 1.0f : 0.0f;
            // M[k][c] = diff at bin base + c*16 + k  (column c = one 16-bin segment)
            const int bin0 = base + c * 16 + k0;
            b.x = (float)g_tp[bin0]     * invP - (float)g_fp[bin0]     * invN;
            b.y = (float)g_tp[bin0 + 1] * invP - (float)g_fp[bin0 + 1] * invN;
            acc = __builtin_amdgcn_wmma_f32_16x16x4_f32(
                false, a, false, b, (short)0, acc, false, false);
        }
        // acc[i] (lane half h) = intra-column inclusive scan D[i+8h][c].
        // Column sums D[15][j] live in lane 16+j, acc[7]. Prefix them across columns.
        const float last = acc[7];
        float tot = 0.0f, pref = 0.0f;
        for (int j = 0; j < 16; ++j) {
            float v = __shfl(last, 16 + j);
            if (j < c) pref += v;
            tot += v;
        }
        const float colcarry = carry + pref;
#pragma unroll
        for (int i = 0; i < 8; ++i) {
            float v  = acc[i] + colcarry;    // global cumsum at bin base+c*16+i+8*half
            float av = fabsf(v);
            mx = av > mx ? av : mx;          // padded bins repeat bin-10000 value: harmless
        }
        carry += tot;
    }
    for (int off = 16; off; off >>= 1) {
        float o = __shfl_xor(mx, off);
        mx = o > mx ? o : mx;
    }
    if (lane == 0) out[0] = mx;
}

extern "C" void kernel_launch(void* const* d_in, const int* in_sizes, int n_in,
                              void* d_out, int out_size, void* d_ws, size_t ws_size,
                              hipStream_t stream) {
    const float* preds   = (const float*)d_in[0];
    const float* targets = (const float*)d_in[1];
    const int n = in_sizes[0];

    unsigned* g_tp = (unsigned*)d_ws;
    unsigned* g_fp = g_tp + PADBINS;

    ks_zero<<<(2 * PADBINS + 255) / 256, 256, 0, stream>>>(g_tp);

    // cap elements/block at 16384 so packed 16-bit LDS counters cannot overflow
    int blocks = (n + 16383) / 16384;
    if (blocks < 1) blocks = 1;
    if (blocks > 2048) blocks = 2048;
    ks_hist<<<blocks, 256, 0, stream>>>(preds, targets, g_tp, g_fp, n);

    ks_scan<<<1, 32, 0, stream>>>(g_tp, g_fp, (float*)d_out);
}